// Reconstructor_86517821216138
// MI455X (gfx1250) — compile-verified
//
#include <hip/hip_runtime.h>
#include <math.h>

// ---------------------------------------------------------------------------
// Cryo-EM back-projection reconstructor for MI455X (gfx1250, wave32).
//  - rfft2 implemented as two DFT matmul passes on V_WMMA_F32_16X16X4_F32
//    (full f32 precision on the matrix pipes; fftshifts folded into twiddles)
//  - scatter phase: one thread per (slice,pixel), 32 global_atomic_add_f32.
//    Output volume (135 MB) fits in the 192 MB L2, so atomics stay on-chip.
// ---------------------------------------------------------------------------

#define DBOX 256
#define DX   129
#define DXP  144            // 9 tiles of 16 (padded cols)
#define DC   128
#define NB   16             // batch
#define NSYM 4
#define NSL  64             // NB * NSYM slices
#define CHVOL (DBOX*DBOX*DX)   // 8,454,144 floats per output channel
#define SLICE_F (DBOX*DXP)     // 36,864 floats per per-image plane
#define PI2 6.28318530717958647692f
#define SRATE 1.5f

typedef __attribute__((ext_vector_type(2))) float v2f;
typedef __attribute__((ext_vector_type(8))) float v8f;

// workspace layout (float offsets)
#define OFF_COSW1  0                              // [x][c]  256 x 144
#define OFF_SINW1N (OFF_COSW1 + DBOX*DXP)         // -sin, same shape
#define OFF_COST2  (OFF_SINW1N + DBOX*DXP)        // [r][y]  256 x 256
#define OFF_SINT2P (OFF_COST2 + DBOX*DBOX)        // +sin
#define OFF_SINT2N (OFF_SINT2P + DBOX*DBOX)       // -sin
#define OFF_GR     (OFF_SINT2N + DBOX*DBOX)       // [b][y][c] 16 x 256 x 144
#define OFF_GI     (OFF_GR + NB*SLICE_F)
#define OFF_FR     (OFF_GI + NB*SLICE_F)
#define OFF_FI     (OFF_FR + NB*SLICE_F)
#define OFF_ROT    (OFF_FI + NB*SLICE_F)          // 64 x 9
#define OFF_SH     (OFF_ROT + NSL*9)              // 16 x 2

static __device__ __forceinline__ v8f wmma_f32(v2f a, v2f b, v8f c) {
    return __builtin_amdgcn_wmma_f32_16x16x4_f32(
        /*neg_a=*/false, a, /*neg_b=*/false, b,
        /*c_mod=*/(short)0, c, /*reuse_a=*/false, /*reuse_b=*/false);
}

// --------------------------- zero the output -------------------------------
__global__ void k_zero(float4* __restrict__ out, int n4) {
    int i = blockIdx.x * blockDim.x + threadIdx.x;
    int stride = gridDim.x * blockDim.x;
    float4 z = make_float4(0.f, 0.f, 0.f, 0.f);
    for (; i < n4; i += stride) out[i] = z;
}

// --------------------------- twiddle factors -------------------------------
// Pass 1 (x-dir):  W1[x][c]  = exp(-2*pi*i * c * (x - dc) / D)
// Pass 2 (y-dir):  T2[r][y]  = exp(-2*pi*i * (r - dc) * (y - dc) / D)
// (input fftshift + output row fftshift folded into the (…-dc) terms)
__global__ void k_twiddle(float* __restrict__ ws) {
    int i = blockIdx.x * blockDim.x + threadIdx.x;   // 0 .. 65535
    if (i < DBOX * DBOX) {
        int r = i >> 8, y = i & 255;
        float ang = (PI2 / (float)DBOX) * (float)((r - DC) * (y - DC));
        float s, c; sincosf(ang, &s, &c);
        ws[OFF_COST2 + i]  = c;
        ws[OFF_SINT2P + i] = s;
        ws[OFF_SINT2N + i] = -s;
    }
    if (i < DBOX * DXP) {
        int x = i / DXP, cc = i - x * DXP;
        float ang = (PI2 / (float)DBOX) * (float)(cc * (x - DC));
        float s, c; sincosf(ang, &s, &c);
        ws[OFF_COSW1 + i]  = c;
        ws[OFF_SINW1N + i] = -s;
    }
}

// --------------------- rotation matrices + shifts --------------------------
// Replicates the reference's reshape pairing exactly:
//   rot_e = einsum('sij,bjk->sbik').reshape(N,3,3)  -> slice n uses
//   sym[n>>4] @ rotMats[n&15], while data uses image b = n>>2.
__global__ void k_prep(const float* __restrict__ rotMats,
                       const float* __restrict__ hwShift,
                       const float* __restrict__ sym,
                       float* __restrict__ ws) {
    int t = threadIdx.x;
    if (t < NSL) {
        int sp = t >> 4, bp = t & 15;
        const float* S = sym + sp * 9;
        const float* R = rotMats + bp * 9;
        #pragma unroll
        for (int i = 0; i < 3; ++i)
            #pragma unroll
            for (int j = 0; j < 3; ++j) {
                float acc = 0.f;
                #pragma unroll
                for (int k = 0; k < 3; ++k) acc += S[i*3+k] * R[k*3+j];
                ws[OFF_ROT + t*9 + i*3 + j] = acc;
            }
    }
    if (t < NB) {
        ws[OFF_SH + t*2 + 0] = hwShift[t*2 + 0] / SRATE;
        ws[OFF_SH + t*2 + 1] = hwShift[t*2 + 1] / SRATE;
    }
}

// ----------------- pass 1: DFT along x (real -> complex) -------------------
// G[b,y,c] = sum_x img[b,y,x] * W1[x,c].   One wave per 16x16 output tile.
// Tiles: 16 images * 16 row-tiles * 9 col-tiles = 2304; 8 waves/block -> 288.
__global__ void k_dft_x(const float* __restrict__ img, float* __restrict__ ws) {
    int wave = threadIdx.x >> 5;
    int lane = threadIdx.x & 31;
    int tile = blockIdx.x * 8 + wave;
    int b     = tile / (16 * 9);
    int rem   = tile - b * (16 * 9);
    int ytile = rem / 9;
    int ctile = rem - ytile * 9;
    int y_base = ytile << 4, c_base = ctile << 4;
    int lm = lane & 15;     // M (row) for A, N (col) for B/C
    int lh = lane >> 4;     // K half-select

    const float* arow  = img + ((size_t)b * DBOX + (y_base + lm)) * DBOX;
    const float* cosW  = ws + OFF_COSW1;
    const float* sinWn = ws + OFF_SINW1N;
    int n = c_base + lm;

    v8f accR = {}; v8f accI = {};
    for (int k = 0; k < DBOX; k += 4) {
        int x0 = k + (lh << 1);
        v2f a = *(const v2f*)(arow + x0);           // A[M=lm][K=x0, x0+1]
        v2f bc, bs;
        bc.x = cosW[(size_t)x0 * DXP + n];  bc.y = cosW[(size_t)(x0+1) * DXP + n];
        bs.x = sinWn[(size_t)x0 * DXP + n]; bs.y = sinWn[(size_t)(x0+1) * DXP + n];
        accR = wmma_f32(a, bc, accR);
        accI = wmma_f32(a, bs, accI);
    }
    float* Gr = ws + OFF_GR + (size_t)b * SLICE_F;
    float* Gi = ws + OFF_GI + (size_t)b * SLICE_F;
    #pragma unroll
    for (int v = 0; v < 8; ++v) {
        int r = y_base + v + (lh << 3);             // C: M = v + 8*lh
        Gr[(size_t)r * DXP + n] = accR[v];
        Gi[(size_t)r * DXP + n] = accI[v];
    }
}

// -------------- pass 2: DFT along y (complex) + shift phase ----------------
// f[b,r,c] = sum_y T2[r,y] * G[b,y,c];  T2 = C - i*S
//   fr = C*Gr + S*Gi   (A in {cosT2, sinT2p})
//   fi = C*Gi - S*Gr   (A in {cosT2, sinT2n})
__global__ void k_dft_y(float* __restrict__ ws) {
    int wave = threadIdx.x >> 5;
    int lane = threadIdx.x & 31;
    int tile = blockIdx.x * 8 + wave;
    int b     = tile / (16 * 9);
    int rem   = tile - b * (16 * 9);
    int rtile = rem / 9;
    int ctile = rem - rtile * 9;
    int r_base = rtile << 4, c_base = ctile << 4;
    int lm = lane & 15;
    int lh = lane >> 4;

    const float* C2  = ws + OFF_COST2  + (size_t)(r_base + lm) * DBOX;
    const float* S2p = ws + OFF_SINT2P + (size_t)(r_base + lm) * DBOX;
    const float* S2n = ws + OFF_SINT2N + (size_t)(r_base + lm) * DBOX;
    const float* Gr  = ws + OFF_GR + (size_t)b * SLICE_F;
    const float* Gi  = ws + OFF_GI + (size_t)b * SLICE_F;
    int n = c_base + lm;

    v8f fr = {}; v8f fi = {};
    for (int k = 0; k < DBOX; k += 4) {
        int y0 = k + (lh << 1);
        v2f ac  = *(const v2f*)(C2  + y0);
        v2f asp = *(const v2f*)(S2p + y0);
        v2f asn = *(const v2f*)(S2n + y0);
        v2f bgr, bgi;
        bgr.x = Gr[(size_t)y0 * DXP + n]; bgr.y = Gr[(size_t)(y0+1) * DXP + n];
        bgi.x = Gi[(size_t)y0 * DXP + n]; bgi.y = Gi[(size_t)(y0+1) * DXP + n];
        fr = wmma_f32(ac,  bgr, fr);
        fr = wmma_f32(asp, bgi, fr);
        fi = wmma_f32(ac,  bgi, fi);
        fi = wmma_f32(asn, bgr, fi);
    }

    // epilogue: apply exp(-2*pi*i*(ky*sh0 + kx*sh1)) and store
    float sh0 = ws[OFF_SH + b*2 + 0];
    float sh1 = ws[OFF_SH + b*2 + 1];
    float kxv = (float)n * (1.0f / DBOX);
    float* Fr = ws + OFF_FR + (size_t)b * SLICE_F;
    float* Fi = ws + OFF_FI + (size_t)b * SLICE_F;
    #pragma unroll
    for (int v = 0; v < 8; ++v) {
        int r = r_base + v + (lh << 3);
        float kyv = (float)(r - DC) * (1.0f / DBOX);
        float th = PI2 * (kyv * sh0 + kxv * sh1);
        float s, c; sincosf(th, &s, &c);
        float vr = fr[v], vi = fi[v];
        Fr[(size_t)r * DXP + n] = vr * c + vi * s;   // (vr+i*vi)*(c-i*s)
        Fi[(size_t)r * DXP + n] = vi * c - vr * s;
    }
}

// ----------------------- trilinear scatter (splat) -------------------------
// One thread per (slice n, pixel r,c). 8 corners x 4 channels of f32 atomics;
// whole output volume is L2-resident on MI455X.
__global__ void k_splat(const float* __restrict__ ctf,
                        const float* __restrict__ ws,
                        float* __restrict__ out) {
    int n = blockIdx.y;                         // slice 0..63
    int flat = blockIdx.x * blockDim.x + threadIdx.x;   // exactly D*DX threads
    int r = flat / DX;
    int c = flat - r * DX;
    int b = n >> 2;                             // data image index (n = b*4+s)

    const float* R = ws + OFF_ROT + n * 9;
    float kxv = (float)c * (1.0f / DBOX);
    float kyv = (float)(r - DC) * (1.0f / DBOX);

    float fr = ws[OFF_FR + (size_t)b * SLICE_F + (size_t)r * DXP + c];
    float fi = ws[OFF_FI + (size_t)b * SLICE_F + (size_t)r * DXP + c];
    float ct = ctf[((size_t)b * DBOX + r) * DX + c];
    float vr = fr * ct, vi = fi * ct, w3 = ct * ct;

    float c0 = R[0] * kxv + R[1] * kyv;
    float c1 = R[3] * kxv + R[4] * kyv;
    float c2 = R[6] * kxv + R[7] * kyv;
    if (c0 < 0.f) { c0 = -c0; c1 = -c1; c2 = -c2; vi = -vi; }

    float xi = c0 * (float)DBOX;
    float yi = c1 * (float)DBOX + (float)DC;
    float zi = c2 * (float)DBOX + (float)DC;
    float xf = floorf(xi), yf = floorf(yi), zf = floorf(zi);
    float tx = xi - xf, ty = yi - yf, tz = zi - zf;
    int x0 = (int)xf, y0 = (int)yf, z0 = (int)zf;

    #pragma unroll
    for (int dz = 0; dz < 2; ++dz) {
        int z = z0 + dz;
        if ((unsigned)z >= (unsigned)DBOX) continue;
        float wz = dz ? tz : 1.f - tz;
        #pragma unroll
        for (int dy = 0; dy < 2; ++dy) {
            int y = y0 + dy;
            if ((unsigned)y >= (unsigned)DBOX) continue;
            float wy = dy ? ty : 1.f - ty;
            #pragma unroll
            for (int dx = 0; dx < 2; ++dx) {
                int x = x0 + dx;
                if ((unsigned)x >= (unsigned)DX) continue;
                float w = wz * wy * (dx ? tx : 1.f - tx);
                size_t base = ((size_t)z * DBOX + y) * DX + x;
                atomicAdd(out + base,                    vr * w);
                atomicAdd(out + (size_t)CHVOL + base,    vi * w);
                atomicAdd(out + 2*(size_t)CHVOL + base,  w3 * w);
                atomicAdd(out + 3*(size_t)CHVOL + base,  w);
            }
        }
    }
}

// ---------------------------------------------------------------------------
extern "C" void kernel_launch(void* const* d_in, const int* in_sizes, int n_in,
                              void* d_out, int out_size, void* d_ws, size_t ws_size,
                              hipStream_t stream) {
    const float* imgs    = (const float*)d_in[0];   // (16,256,256)
    const float* ctf     = (const float*)d_in[1];   // (16,256,129)
    const float* rotMats = (const float*)d_in[2];   // (16,3,3)
    const float* hwShift = (const float*)d_in[3];   // (16,2)
    const float* sym     = (const float*)d_in[4];   // (4,3,3)
    float* out = (float*)d_out;                     // (4,256,256,129)
    float* ws  = (float*)d_ws;

    k_zero<<<2048, 256, 0, stream>>>((float4*)out, out_size / 4);
    k_twiddle<<<(DBOX * DBOX) / 256, 256, 0, stream>>>(ws);
    k_prep<<<1, 64, 0, stream>>>(rotMats, hwShift, sym, ws);
    k_dft_x<<<288, 256, 0, stream>>>(imgs, ws);     // 2304 tiles, 8 waves/blk
    k_dft_y<<<288, 256, 0, stream>>>(ws);
    dim3 g((DBOX * DX) / 256, NSL);                 // 129 x 64 blocks, exact
    k_splat<<<g, 256, 0, stream>>>(ctf, ws, out);
}